// FunctionConv_43611097924170
// MI455X (gfx1250) — compile-verified
//
#include <hip/hip_runtime.h>

typedef float v2f __attribute__((ext_vector_type(2)));
typedef float v8f __attribute__((ext_vector_type(8)));

#define D 64          // D_IN == D_OUT == 64 in this problem
#define TILE 16       // WMMA M/N

// ---------------------------------------------------------------------------
// Kernel 1: initialize accumulators / buckets (runs every launch -> deterministic)
// ---------------------------------------------------------------------------
__global__ __launch_bounds__(256)
void k_init(float* __restrict__ out, float* __restrict__ deg,
            int* __restrict__ cnt, int* __restrict__ bkt,
            long n64, long n, int T, long bktLen) {
    long i = (long)blockIdx.x * blockDim.x + threadIdx.x;
    if (i < n64) { out[i] = 0.0f; return; }
    i -= n64;
    if (i < n) { deg[i] = 0.0f; return; }
    i -= n;
    if (i < T) { cnt[i] = 0; return; }
    i -= T;
    if (i < bktLen) bkt[i] = -1;
}

// ---------------------------------------------------------------------------
// Kernel 2: edge scatter-add. 32 lanes per edge, float2 per lane.
// Gathered reads are coalesced (256B per edge); atomics land in L2 (25.6MB
// accumulator << 192MB L2).
// ---------------------------------------------------------------------------
__global__ __launch_bounds__(256)
void k_edges(const float* __restrict__ feat, const int* __restrict__ src,
             const int* __restrict__ dst, float* __restrict__ sum,
             float* __restrict__ deg, int E) {
    int tid  = blockIdx.x * blockDim.x + threadIdx.x;
    int e    = tid >> 5;
    int lane = tid & 31;
    if (e >= E) return;
    int s = src[e];
    int d = dst[e];
    const float2* frow = (const float2*)(feat + (size_t)s * D);
    float2 v = frow[lane];
    float* base = sum + (size_t)d * D + lane * 2;
    atomicAdd(base,     v.x);
    atomicAdd(base + 1, v.y);
    if (lane == 0) atomicAdd(deg + d, 1.0f);
}

// ---------------------------------------------------------------------------
// Kernel 3: neigh = sum / max(deg,1)  (in place in d_out) + type histogram
// ---------------------------------------------------------------------------
__global__ __launch_bounds__(256)
void k_finalize(float* __restrict__ out, const float* __restrict__ deg,
                const int* __restrict__ ntype, int* __restrict__ cnt, long n64) {
    long i = (long)blockIdx.x * blockDim.x + threadIdx.x;
    if (i >= n64) return;
    int node = (int)(i >> 6);
    float g  = deg[node];
    out[i] *= 1.0f / fmaxf(g, 1.0f);
    if ((i & 63) == 0) atomicAdd(cnt + ntype[node], 1);
}

// ---------------------------------------------------------------------------
// Kernel 4: 16-aligned exclusive prefix sum over T(=16) type counts.
// cur[t] = start slot of type-t bucket (tile aligned -> tiles are type-pure).
// ---------------------------------------------------------------------------
__global__ void k_offsets(const int* __restrict__ cnt, int* __restrict__ cur, int T) {
    if (threadIdx.x == 0 && blockIdx.x == 0) {
        int running = 0;
        for (int t = 0; t < T; ++t) {
            cur[t] = running;
            running += (cnt[t] + (TILE - 1)) & ~(TILE - 1);
        }
    }
}

// ---------------------------------------------------------------------------
// Kernel 5: scatter node ids into type buckets
// ---------------------------------------------------------------------------
__global__ __launch_bounds__(256)
void k_scatter(const int* __restrict__ ntype, int* __restrict__ cur,
               int* __restrict__ bkt, int n) {
    int i = blockIdx.x * blockDim.x + threadIdx.x;
    if (i >= n) return;
    int t   = ntype[i];
    int pos = atomicAdd(cur + t, 1);
    bkt[pos] = i;
}

// ---------------------------------------------------------------------------
// Kernel 6: one wave per 16-node tile; rst = neigh @ W[t] + b[t] via
// V_WMMA_F32_16X16X4_F32, in place over d_out.
//
// F32 WMMA fragment layout (ISA 7.12.2):
//   A (16x4):  lanes 0-15 row M=lane, K={0,1}; lanes 16-31 row M=lane-16, K={2,3}
//   B (4x16):  lanes 0-15 col N=lane, K={0,1}; lanes 16-31 col N=lane-16, K={2,3}
//   C (16x16): VGPR j -> row M=j (lanes<16) / M=8+j (lanes>=16), col N=lane&15
// ---------------------------------------------------------------------------
__global__ __launch_bounds__(256)
void k_gemm(float* __restrict__ out, const float* __restrict__ W,
            const float* __restrict__ bias, const int* __restrict__ ntype,
            const int* __restrict__ bkt, int nTiles) {
    int wave = (blockIdx.x * blockDim.x + threadIdx.x) >> 5;  // wave-uniform
    int lane = threadIdx.x & 31;
    if (wave >= nTiles) return;                                // uniform exit
    int base = wave * TILE;

    int n0 = bkt[base];
    if (n0 < 0) return;                 // fully-padded tile: uniform exit
    int t = ntype[n0];                  // tile is type-pure by construction

    // ---- load the whole 16x64 A fragment into registers (32 VGPRs/lane) ----
    int rowA  = lane & 15;
    int nodeA = bkt[base + rowA];
    int klo   = (lane < 16) ? 0 : 2;
    float a[2 * TILE];
    const float* arow = (nodeA >= 0) ? (out + (size_t)nodeA * D) : out;
    #pragma unroll
    for (int s = 0; s < 16; ++s) {
        float x0 = arow[4 * s + klo];
        float x1 = arow[4 * s + klo + 1];
        a[2 * s]     = (nodeA >= 0) ? x0 : 0.0f;
        a[2 * s + 1] = (nodeA >= 0) ? x1 : 0.0f;
    }

    // node ids for the 8 C rows this lane stores
    int mbase = (lane < 16) ? 0 : 8;
    int nodeM[8];
    #pragma unroll
    for (int j = 0; j < 8; ++j) nodeM[j] = bkt[base + mbase + j];

    const float* Wt = W + (size_t)t * D * D;   // [64 x 64], row = k, col = o
    int col = lane & 15;

    #pragma unroll
    for (int ct = 0; ct < 4; ++ct) {           // 4 output column tiles of 16
        v8f c = {0.f, 0.f, 0.f, 0.f, 0.f, 0.f, 0.f, 0.f};
        #pragma unroll
        for (int s = 0; s < 16; ++s) {         // K = 64 in steps of 4
            v2f av; av.x = a[2 * s]; av.y = a[2 * s + 1];
            v2f bv;
            bv.x = Wt[(size_t)(4 * s + klo)     * D + ct * 16 + col];
            bv.y = Wt[(size_t)(4 * s + klo + 1) * D + ct * 16 + col];
            c = __builtin_amdgcn_wmma_f32_16x16x4_f32(
                    /*neg_a=*/false, av, /*neg_b=*/false, bv,
                    /*c_mod=*/(short)0, c, /*reuse_a=*/false, /*reuse_b=*/false);
        }
        float bv = bias[t * D + ct * 16 + col];  // one column per lane
        #pragma unroll
        for (int j = 0; j < 8; ++j) {
            if (nodeM[j] >= 0)
                out[(size_t)nodeM[j] * D + ct * 16 + col] = c[j] + bv;
        }
    }
}

// ---------------------------------------------------------------------------
extern "C" void kernel_launch(void* const* d_in, const int* in_sizes, int n_in,
                              void* d_out, int out_size, void* d_ws, size_t ws_size,
                              hipStream_t stream) {
    const float* feat  = (const float*)d_in[0];
    const float* gateW = (const float*)d_in[1];
    const float* gateB = (const float*)d_in[2];
    const int*   src   = (const int*)d_in[3];
    const int*   dst   = (const int*)d_in[4];
    const int*   ntype = (const int*)d_in[5];
    // d_in[6] = act_flag: activation is None in this config -> unused.

    const int n = in_sizes[0] / D;            // 100,000
    const int E = in_sizes[3];                // 1,000,000
    const int T = in_sizes[1] / (D * D);      // 16

    float* out = (float*)d_out;               // doubles as sum/neigh accumulator

    // workspace layout (4B words): deg[n] | cnt[T] | cur[T] | bkt[bktLen]
    float* deg = (float*)d_ws;
    int*   cnt = (int*)d_ws + n;
    int*   cur = cnt + T;
    int*   bkt = cur + T;
    const long bktLen = (((long)n + (long)TILE * T + TILE - 1) / TILE) * TILE;
    const int  nTiles = (int)(bktLen / TILE);

    const long n64 = (long)n * D;

    // 1) init accumulators/buckets
    {
        long total = n64 + n + T + bktLen;
        long grid  = (total + 255) / 256;
        k_init<<<(unsigned)grid, 256, 0, stream>>>(out, deg, cnt, bkt, n64, n, T, bktLen);
    }
    // 2) edge scatter-add (32 lanes / edge)
    {
        long total = (long)E * 32;
        long grid  = (total + 255) / 256;
        k_edges<<<(unsigned)grid, 256, 0, stream>>>(feat, src, dst, out, deg, E);
    }
    // 3) mean + type histogram
    {
        long grid = (n64 + 255) / 256;
        k_finalize<<<(unsigned)grid, 256, 0, stream>>>(out, deg, ntype, cnt, n64);
    }
    // 4) tile-aligned bucket offsets
    k_offsets<<<1, 1, 0, stream>>>(cnt, cur, T);
    // 5) bucket scatter
    k_scatter<<<(n + 255) / 256, 256, 0, stream>>>(ntype, cur, bkt, n);
    // 6) typed GEMM via f32 WMMA (8 waves / block, one tile per wave)
    k_gemm<<<(nTiles + 7) / 8, 256, 0, stream>>>(out, gateW, gateB, ntype, bkt, nTiles);
}